// RFNTModel_61838939128058
// MI455X (gfx1250) — compile-verified
//
#include <hip/hip_runtime.h>
#include <math.h>

// ---------------------------------------------------------------------------
// dca_similarity for MI455X (gfx1250, wave32)
//
// out[b,i,j] = (x1[b,i]-m1[b]) * (x2[b,j]-m2[b]) / ((D-1)*std1[b]*std2[b])
//
// B=64, D=1024. Output = 64*1024*1024 fp32 = 256 MB -> pure store-BW bound
// (~11.5 us at 23.3 TB/s). Strategy:
//   Pass 1: per-batch stats; write a1=(x1-m1)*s, a2=(x2-m2) to workspace.
//   Pass 2: rank-1 GEMM tiles via V_WMMA_F32_16X16X4_F32 (K=0 column/row
//           carry the vectors, K=1..3 zero), streamed out with NT stores.
//   All loads are unconditional ((lane&15) indexing + value select) so the
//   backend emits straight-line load clauses instead of exec-branches.
// ---------------------------------------------------------------------------

typedef __attribute__((ext_vector_type(2))) float v2f;
typedef __attribute__((ext_vector_type(8))) float v8f;

#define DIM 1024
#define BATCH 64

// ---------------- Pass 1: per-batch mean / scale + centered vectors --------
__global__ void dca_stats_kernel(const float* __restrict__ x1,
                                 const float* __restrict__ x2,
                                 float* __restrict__ a1,
                                 float* __restrict__ a2) {
    const int b   = blockIdx.x;      // one block per batch row
    const int tid = threadIdx.x;     // 256 threads -> 4 elems each
    const float* p1 = x1 + b * DIM;
    const float* p2 = x2 + b * DIM;

    float v1[4], v2l[4];
    float s1 = 0.f, s2 = 0.f, q1 = 0.f, q2 = 0.f;
#pragma unroll
    for (int k = 0; k < 4; ++k) {
        float u1 = p1[tid + k * 256];
        float u2 = p2[tid + k * 256];
        v1[k] = u1; v2l[k] = u2;
        s1 += u1; s2 += u2;
        q1 += u1 * u1; q2 += u2 * u2;
    }

    // wave32 butterfly reduction
#pragma unroll
    for (int off = 16; off > 0; off >>= 1) {
        s1 += __shfl_down(s1, off, 32);
        s2 += __shfl_down(s2, off, 32);
        q1 += __shfl_down(q1, off, 32);
        q2 += __shfl_down(q2, off, 32);
    }

    __shared__ float sm[4][8];
    __shared__ float stats[3];       // m1, m2, s
    const int lane = tid & 31;
    const int wave = tid >> 5;       // 8 waves per block (wave32)
    if (lane == 0) {
        sm[0][wave] = s1; sm[1][wave] = s2;
        sm[2][wave] = q1; sm[3][wave] = q2;
    }
    __syncthreads();
    if (tid == 0) {
        float S1 = 0.f, S2 = 0.f, Q1 = 0.f, Q2 = 0.f;
#pragma unroll
        for (int w = 0; w < 8; ++w) {
            S1 += sm[0][w]; S2 += sm[1][w];
            Q1 += sm[2][w]; Q2 += sm[3][w];
        }
        const float m1 = S1 * (1.0f / DIM);
        const float m2 = S2 * (1.0f / DIM);
        const float var1 = (Q1 - DIM * m1 * m1) * (1.0f / (DIM - 1));
        const float var2 = (Q2 - DIM * m2 * m2) * (1.0f / (DIM - 1));
        stats[0] = m1;
        stats[1] = m2;
        stats[2] = 1.0f / ((float)(DIM - 1) * sqrtf(var1 * var2));
    }
    __syncthreads();
    const float m1 = stats[0], m2 = stats[1], s = stats[2];
#pragma unroll
    for (int k = 0; k < 4; ++k) {
        a1[b * DIM + tid + k * 256] = (v1[k] - m1) * s;  // fold full scale into a1
        a2[b * DIM + tid + k * 256] = (v2l[k] - m2);
    }
}

// ---------------- Pass 2: WMMA rank-1 outer-product tiles ------------------
// Grid: BATCH * (DIM/16) blocks; block = 256 threads = 8 waves.
// Each block emits one 16-row x 1024-col strip of one batch (64 KB).
// Each wave: 8 N-tiles of 16 (128 cols); B segments preloaded as one clause.
__global__ void dca_outer_wmma_kernel(const float* __restrict__ a1,
                                      const float* __restrict__ a2,
                                      float* __restrict__ out) {
    const int strip = blockIdx.x;            // 0 .. BATCH*64-1
    const int b     = strip >> 6;            // batch
    const int m0    = (strip & 63) << 4;     // row-tile base
    const int lane  = threadIdx.x & 31;
    const int wave  = threadIdx.x >> 5;      // 0..7

    const bool lo    = lane < 16;
    const int  nlane = lane & 15;            // in-bounds index for ALL lanes
    const int  n_base = wave * 128;          // 8 waves x 128 cols = 1024

    const float* a2b = a2 + b * DIM;

    // Unconditional loads (lanes 16-31 re-read lanes 0-15's addresses ->
    // broadcast, L2-resident). Zeroing is done with a value select, so the
    // backend emits one load clause, no exec branches.
    float bseg[8];
#pragma unroll
    for (int t = 0; t < 8; ++t) {
        bseg[t] = a2b[n_base + t * 16 + nlane];
    }
    const float aseg = a1[b * DIM + m0 + nlane];

    // A matrix (16x4 f32): lanes 0-15 hold {K=0,K=1} in VGPR0/1,
    // lanes 16-31 hold {K=2,K=3}. a1 segment in K=0, zero the rest.
    v2f A;
    A.x = lo ? aseg : 0.0f;
    A.y = 0.0f;

    float* outb = out + (size_t)b * DIM * DIM + (size_t)m0 * DIM;
    const int mhalf = lo ? 0 : 8;            // C/D row split across lane halves

#pragma unroll
    for (int t = 0; t < 8; ++t) {
        const int n0 = n_base + t * 16;
        // B matrix (4x16 f32): row K striped across lanes within a VGPR;
        // VGPR0 = rows {K=0 (lanes 0-15), K=2 (lanes 16-31)}, VGPR1 = {K=1,K=3}.
        // a2 segment in K=0 row, zeros elsewhere -> D = a1 (outer) a2.
        v2f Bm;
        Bm.x = lo ? bseg[t] : 0.0f;
        Bm.y = 0.0f;

        v8f C = {};
        // 8 args: (neg_a, A, neg_b, B, c_mod, C, reuse_a, reuse_b)
        v8f Dm = __builtin_amdgcn_wmma_f32_16x16x4_f32(
            false, A, false, Bm, (short)0, C, false, false);

        // C/D layout: VGPR r -> row m0+r (lanes 0-15) / m0+r+8 (lanes 16-31),
        // col = n0 + (lane & 15). Stream out with non-temporal hint: 256 MB
        // output > 192 MB L2 and is never re-read.
        float* dst = outb + (size_t)mhalf * DIM + n0 + nlane;
#pragma unroll
        for (int r = 0; r < 8; ++r) {
            __builtin_nontemporal_store(Dm[r], dst + (size_t)r * DIM);
        }
    }
}

// ---------------------------------------------------------------------------
extern "C" void kernel_launch(void* const* d_in, const int* in_sizes, int n_in,
                              void* d_out, int out_size, void* d_ws, size_t ws_size,
                              hipStream_t stream) {
    (void)in_sizes; (void)n_in; (void)out_size; (void)ws_size;

    const float* x1 = (const float*)d_in[0];
    const float* x2 = (const float*)d_in[1];
    float* out = (float*)d_out;

    float* a1 = (float*)d_ws;            // BATCH*DIM floats (scaled centered x1)
    float* a2 = a1 + BATCH * DIM;        // BATCH*DIM floats (centered x2)

    dca_stats_kernel<<<BATCH, 256, 0, stream>>>(x1, x2, a1, a2);
    dca_outer_wmma_kernel<<<BATCH * (DIM / 16), 256, 0, stream>>>(a1, a2, out);
}